// Deep_AD_F_58213986730479
// MI455X (gfx1250) — compile-verified
//
#include <hip/hip_runtime.h>
#include <hip/hip_bf16.h>

#define HH 512
#define WW 512
#define TILE 64
#define HALO 6
#define TSZ 76               // TILE + 2*HALO
#define NPLANE (TSZ * TSZ)   // 5776

typedef unsigned int v4u __attribute__((ext_vector_type(4)));
typedef int v4i __attribute__((ext_vector_type(4)));
typedef int v8i __attribute__((ext_vector_type(8)));

__global__ __launch_bounds__(256) void diffuse3_kernel(
    const float* __restrict__ x, const float* __restrict__ Wt,
    const float* __restrict__ bias, float* __restrict__ out)
{
    __shared__ float plane[2][NPLANE];
    __shared__ float sW[448];   // [0..431] = W (3*4*4*3*3), [432..443] = b

    const int tid = threadIdx.x;
    const int n   = blockIdx.z;
    const int gy0 = blockIdx.y * TILE;
    const int gx0 = blockIdx.x * TILE;
    const float* xin = x + (size_t)n * HH * WW;

    // ---- Tile load: gfx1250 async global->LDS; zero-fill halo outside image.
    for (int i = tid; i < NPLANE; i += 256) {
        int ly = i / TSZ;
        int lx = i - ly * TSZ;
        int gy = gy0 - HALO + ly;
        int gx = gx0 - HALO + lx;
        if ((unsigned)gy < (unsigned)HH && (unsigned)gx < (unsigned)WW) {
            unsigned loff = (unsigned)(unsigned long long)(&plane[0][i]);
            unsigned long long ga = (unsigned long long)(xin + (size_t)gy * WW + gx);
            asm volatile("global_load_async_to_lds_b32 %0, %1, off"
                         :: "v"(loff), "v"(ga) : "memory");
        } else {
            plane[0][i] = 0.0f;
        }
    }

    // ---- Weights: TDM DMA of the 432-float W tensor into LDS (wave 0 only).
    if (tid < 32) {
        unsigned ldsW = (unsigned)(unsigned long long)(&sW[0]);
        unsigned long long ga = (unsigned long long)Wt;
        v4u g0;
        g0[0] = 1u;                                    // count=1 (valid descriptor)
        g0[1] = ldsW;                                  // lds_addr (bytes)
        g0[2] = (unsigned)(ga & 0xffffffffull);        // global_addr[31:0]
        g0[3] = (unsigned)((ga >> 32) & 0x01ffffffull) // global_addr[56:32]
              | (2u << 30);                            // type=2 ("image")
        v8i g1;
        g1[0] = (int)(2u << 16);     // data_size=2 (4B); wg_mask=0
        g1[1] = (int)(432u << 16);   // tensor_dim0[15:0]=432 at bits 63:48
        g1[2] = (int)(1u << 16);     // tensor_dim0 hi=0 | tensor_dim1=1 at bits 95:80
        g1[3] = (int)(432u << 16);   // tensor_dim1 hi=0 | tile_dim0=432
        g1[4] = 1;                   // tile_dim1=1, tile_dim2=0
        g1[5] = 432;                 // tensor_dim0_stride[31:0]
        g1[6] = (int)(432u << 16);   // tensor_dim1_stride[15:0]
        g1[7] = 0;
        v4i g2; g2[0] = 0; g2[1] = 0; g2[2] = 0; g2[3] = 0;
        v4i g3 = g2;
#if defined(__clang_major__) && (__clang_major__ >= 23)
        v8i g4; g4[0]=0; g4[1]=0; g4[2]=0; g4[3]=0; g4[4]=0; g4[5]=0; g4[6]=0; g4[7]=0;
        __builtin_amdgcn_tensor_load_to_lds(g0, g1, g2, g3, g4, 0);
#else
        __builtin_amdgcn_tensor_load_to_lds(g0, g1, g2, g3, 0);
#endif
        __builtin_amdgcn_s_wait_tensorcnt(0);
    } else if (tid >= 32 && tid < 44) {
        sW[432 + (tid - 32)] = bias[tid - 32];   // 12 biases via plain loads
    }

    asm volatile("s_wait_asynccnt 0" ::: "memory");
    __syncthreads();

    const int tx = tid & 15, ty = tid >> 4;

    for (int it = 0; it < 3; ++it) {
        const float* src = plane[it & 1];
        float* dst       = plane[(it & 1) ^ 1];
        const float* Wb  = &sW[it * 144];

        // Collapsed 21-tap stencil weights (shift + 3x3 composed), in registers.
        float wc[4][5][5];
        #pragma unroll
        for (int k = 0; k < 4; ++k)
            #pragma unroll
            for (int a = 0; a < 5; ++a)
                #pragma unroll
                for (int c = 0; c < 5; ++c) wc[k][a][c] = 0.f;
        #pragma unroll
        for (int k = 0; k < 4; ++k) {
            #pragma unroll
            for (int dy = 0; dy < 3; ++dy) {
                #pragma unroll
                for (int dx = 0; dx < 3; ++dx) {
                    wc[k][dy    ][dx + 1] += Wb[k*36 + 0*9 + dy*3 + dx]; // north
                    wc[k][dy + 2][dx + 1] += Wb[k*36 + 1*9 + dy*3 + dx]; // south
                    wc[k][dy + 1][dx    ] += Wb[k*36 + 2*9 + dy*3 + dx]; // west
                    wc[k][dy + 1][dx + 2] += Wb[k*36 + 3*9 + dy*3 + dx]; // east
                }
            }
        }
        float bb[4];
        #pragma unroll
        for (int k = 0; k < 4; ++k) bb[k] = sW[432 + it*4 + k];

        const int  off  = 2 + 2*it;      // region top-left in tile coords
        const int  R    = 72 - 4*it;     // 72, 68, 64
        const bool last = (it == 2);

        for (int yy = ty; yy < R; yy += 16) {
            const int ly = off + yy;
            const int gy = gy0 - HALO + ly;
            for (int xx = tx; xx < R; xx += 16) {
                const int lx = off + xx;
                const int gx = gx0 - HALO + lx;

                float pz[5][5];
                #pragma unroll
                for (int a = 0; a < 5; ++a) {
                    #pragma unroll
                    for (int c = 0; c < 5; ++c) {
                        if ((a == 0 || a == 4) && (c == 0 || c == 4)) { pz[a][c] = 0.f; continue; }
                        pz[a][c] = src[(ly - 2 + a) * TSZ + (lx - 2 + c)];
                    }
                }

                float acc[4];
                #pragma unroll
                for (int k = 0; k < 4; ++k) acc[k] = bb[k];

                const bool border = (gy == 0) | (gy == HH-1) | (gx == 0) | (gx == WW-1);
                if (!border) {
                    // interior: collapsed 21-tap stencil, 84 FMA
                    #pragma unroll
                    for (int a = 0; a < 5; ++a) {
                        #pragma unroll
                        for (int c = 0; c < 5; ++c) {
                            if ((a == 0 || a == 4) && (c == 0 || c == 4)) continue;
                            const float v = pz[a][c];
                            #pragma unroll
                            for (int k = 0; k < 4; ++k)
                                acc[k] = fmaf(wc[k][a][c], v, acc[k]);
                        }
                    }
                } else {
                    // exact two-stage form with conv SAME-pad mask at the image edge
                    #pragma unroll
                    for (int dy = 0; dy < 3; ++dy) {
                        #pragma unroll
                        for (int dx = 0; dx < 3; ++dx) {
                            const int qy = gy + dy - 1, qx = gx + dx - 1;
                            const float m = ((unsigned)qy < (unsigned)HH &&
                                             (unsigned)qx < (unsigned)WW) ? 1.f : 0.f;
                            const float vn = pz[dy    ][dx + 1] * m;
                            const float vs = pz[dy + 2][dx + 1] * m;
                            const float vw = pz[dy + 1][dx    ] * m;
                            const float ve = pz[dy + 1][dx + 2] * m;
                            #pragma unroll
                            for (int k = 0; k < 4; ++k) {
                                float a2 = acc[k];
                                a2 = fmaf(Wb[k*36 + 0*9 + dy*3 + dx], vn, a2);
                                a2 = fmaf(Wb[k*36 + 1*9 + dy*3 + dx], vs, a2);
                                a2 = fmaf(Wb[k*36 + 2*9 + dy*3 + dx], vw, a2);
                                a2 = fmaf(Wb[k*36 + 3*9 + dy*3 + dx], ve, a2);
                                acc[k] = a2;
                            }
                        }
                    }
                }

                float s = 0.f;
                #pragma unroll
                for (int k = 0; k < 4; ++k) {
                    const float d = acc[k];
                    s = fmaf(__expf(-d * d), d, s);    // exp(-d^2)*d, accumulated
                }
                const float val = fmaf(-0.25f, s, pz[2][2]);

                if (last) {
                    out[((size_t)n * HH + gy) * WW + gx] = val;
                } else {
                    const bool inimg = ((unsigned)gy < (unsigned)HH) &
                                       ((unsigned)gx < (unsigned)WW);
                    dst[ly * TSZ + lx] = inimg ? val : 0.f;
                }
            }
        }
        __syncthreads();
    }
}

extern "C" void kernel_launch(void* const* d_in, const int* in_sizes, int n_in,
                              void* d_out, int out_size, void* d_ws, size_t ws_size,
                              hipStream_t stream) {
    const float* x = (const float*)d_in[0];   // [32,1,512,512]
    const float* W = (const float*)d_in[1];   // [3,4,4,3,3]
    const float* b = (const float*)d_in[2];   // [3,4]
    float* out = (float*)d_out;
    dim3 grid(WW / TILE, HH / TILE, 32);
    diffuse3_kernel<<<grid, dim3(256, 1, 1), 0, stream>>>(x, W, b, out);
}